// RoIPooling_2190433321274
// MI455X (gfx1250) — compile-verified
//
#include <hip/hip_runtime.h>
#include <math.h>

// ---------------- problem constants (fixed by the reference) ----------------
#define OUT_DIM 7
#define FH 50
#define FW 50
#define FC 512
#define IMG_H 800
#define IMG_W 800
#define KH (FH / OUT_DIM + 3)   // 10
#define KW (FW / OUT_DIM + 3)   // 10

// ---------------- tiling ----------------
#define CH   8     // channels per workgroup (one TDM 3D tile)
#define MAXD 32    // defensive cap on RoI footprint (in-spec max is 24)

typedef unsigned int v4u __attribute__((ext_vector_type(4)));
typedef int          v4i __attribute__((ext_vector_type(4)));
typedef int          v8i __attribute__((ext_vector_type(8)));

__global__ __launch_bounds__(256) void roi_pool_tdm_kernel(
    const float* __restrict__ x,      // [FC, FH, FW]
    const float* __restrict__ rois,   // [N, 4]
    float* __restrict__ out)          // [N, FC, 7, 7]
{
    __shared__ float tile[CH * MAXD * MAXD];   // 32 KB

    const int blocks_per_roi = FC / CH;        // 64
    const int n  = blockIdx.x / blocks_per_roi;
    const int c0 = (blockIdx.x % blocks_per_roi) * CH;

    // ---- RoI parameters (uniform across the block) ----
    const float wscale = (float)FW / (float)IMG_W;
    const float hscale = (float)FH / (float)IMG_H;
    int rx = (int)(rois[n * 4 + 0] * wscale);
    int ry = (int)(rois[n * 4 + 1] * hscale);
    int rw = (int)(rois[n * 4 + 2] * wscale);
    int rh = (int)(rois[n * 4 + 3] * hscale);
    int sh = rh + 1;
    int sw = rw + 1;
    // defensive clamps (never active for in-spec inputs)
    rx = min(max(rx, 0), FW - 1);
    ry = min(max(ry, 0), FH - 1);
    const int sh_l = min(min(sh, MAXD), FH - ry);   // rows resident in LDS
    const int sw_l = min(min(sw, MAXD), FW - rx);   // cols resident in LDS

#if __has_builtin(__builtin_amdgcn_tensor_load_to_lds)
    // ---- One 3D TDM tile: (sw_l x sh_l x CH) from global -> packed LDS ----
    if (threadIdx.x < 32) {   // one wave issues the DMA
        unsigned long long gaddr =
            (unsigned long long)(const void*)(x + (size_t)c0 * (FH * FW)
                                                + (size_t)ry * FW + rx);
        unsigned int laddr = (unsigned int)(unsigned long long)(void*)tile;

        // D# group 0: count=1 | lds_addr | global_addr[56:0] | type=2
        v4u g0;
        g0[0] = 1u;
        g0[1] = laddr;
        g0[2] = (unsigned int)gaddr;
        g0[3] = (unsigned int)(gaddr >> 32) | (2u << 30);

        // D# group 1
        const unsigned int td0 = (unsigned int)sw_l;   // tensor_dim0
        const unsigned int td1 = (unsigned int)sh_l;   // tensor_dim1
        const unsigned long long s0 = (unsigned long long)FW;        // dim0 stride
        const unsigned long long s1 = (unsigned long long)(FH * FW); // dim1 stride
        v8i g1;
        g1[0] = (int)(2u << 16);                                  // data_size = 4B
        g1[1] = (int)((td0 & 0xFFFFu) << 16);                     // tensor_dim0 lo
        g1[2] = (int)((td0 >> 16) | ((td1 & 0xFFFFu) << 16));     // td0 hi | td1 lo
        g1[3] = (int)((td1 >> 16) | ((unsigned int)sw_l << 16));  // td1 hi | tile_dim0
        g1[4] = (int)((unsigned int)sh_l | ((unsigned int)CH << 16)); // tile1|tile2
        g1[5] = (int)(unsigned int)s0;                            // dim0_stride lo32
        g1[6] = (int)(((unsigned int)(s0 >> 32) & 0xFFFFu) |
                      (((unsigned int)s1 & 0xFFFFu) << 16));      // s0 hi | s1 lo16
        g1[7] = (int)(unsigned int)(s1 >> 16);                    // s1 [47:16]

        // D# group 2: tensor_dim2 = CH, no dim3, no iteration
        v4i g2; g2[0] = CH; g2[1] = 0; g2[2] = 0; g2[3] = 0;
        // D# group 3: unused dims
        v4i g3; g3[0] = 0; g3[1] = 0; g3[2] = 0; g3[3] = 0;
        // 6-arg toolchain variant: extra int32x8 group (zeroed)
        v8i g4 = {0, 0, 0, 0, 0, 0, 0, 0};

        __builtin_amdgcn_tensor_load_to_lds(g0, g1, g2, g3, g4, 0);
        __builtin_amdgcn_s_wait_tensorcnt(0);
    }
    __syncthreads();
#else
    // Cooperative fallback: stage the RoI footprint into LDS with plain loads.
    {
        const int total = CH * sh_l * sw_l;
        for (int i = threadIdx.x; i < total; i += blockDim.x) {
            const int c  = i / (sh_l * sw_l);
            const int r  = i % (sh_l * sw_l);
            const int yy = r / sw_l;
            const int xx = r % sw_l;
            tile[i] = x[(size_t)(c0 + c) * (FH * FW)
                        + (size_t)(ry + yy) * FW + (rx + xx)];
        }
        __syncthreads();
    }
#endif

    // ---- Compute 8 channels x 7x7 bin maxima from LDS ----
    const int total = CH * OUT_DIM * OUT_DIM;   // 392
    for (int idx = threadIdx.x; idx < total; idx += blockDim.x) {
        const int cl = idx / (OUT_DIM * OUT_DIM);
        const int p  = idx % (OUT_DIM * OUT_DIM);
        const int ph = p / OUT_DIM;
        const int pw = p % OUT_DIM;

        // bin bounds relative to (ry, rx), exactly as the reference
        const int ys = (ph * sh) / OUT_DIM;
        int       ye = ((ph + 1) * sh + OUT_DIM - 1) / OUT_DIM;
        const int xs = (pw * sw) / OUT_DIM;
        int       xe = ((pw + 1) * sw + OUT_DIM - 1) / OUT_DIM;
        ye = min(ye, ys + KH);   // reference's Kh truncation (never binds)
        xe = min(xe, xs + KW);   // reference's Kw truncation (never binds)
        ye = min(ye, sh_l);      // stay inside the staged region
        xe = min(xe, sw_l);

        float m = -INFINITY;
        const float* base = &tile[cl * sh_l * sw_l];
        for (int yy = ys; yy < ye; ++yy) {
            const float* row = base + yy * sw_l;
            for (int xx = xs; xx < xe; ++xx)
                m = fmaxf(m, row[xx]);
        }

        // output offset is contiguous in idx -> fully coalesced store
        out[((size_t)n * FC + (size_t)(c0 + cl)) * (OUT_DIM * OUT_DIM) + p] = m;
    }
}

extern "C" void kernel_launch(void* const* d_in, const int* in_sizes, int n_in,
                              void* d_out, int out_size, void* d_ws, size_t ws_size,
                              hipStream_t stream) {
    const float* x    = (const float*)d_in[0];   // (1, 512, 50, 50) fp32
    // d_in[1] = img, only its shape (800x800) matters; hardcoded above
    const float* rois = (const float*)d_in[2];   // (256, 4) fp32
    float* out = (float*)d_out;                  // (256, 512, 7, 7) fp32

    const int n_rois = in_sizes[2] / 4;
    dim3 grid(n_rois * (FC / CH));               // 256 * 64 = 16384 blocks
    dim3 block(256);
    roi_pool_tdm_kernel<<<grid, block, 0, stream>>>(x, rois, out);
    (void)n_in; (void)out_size; (void)d_ws; (void)ws_size;
}